// KuramotoAGM_37512244363631
// MI455X (gfx1250) — compile-verified
//
#include <hip/hip_runtime.h>
#include <math.h>

// Kuramoto coupling via trig identity -> two fp32 GEMMs on the WMMA pipe.
//   S[b,i] = sum_j A[i,j]*sin(p[b,j]),  C[b,i] = sum_j A[i,j]*cos(p[b,j])
//   coupling[b,i] = cos(p[b,i])*S[b,i] - sin(p[b,i])*C[b,i]
// Shapes: batch B=32, dim=1024, T=100. Output (32, 1025).

#define BATCH 32
#define DIM   1024
#define OUTW  1025
#define DT_   0.01f

typedef float v2f __attribute__((ext_vector_type(2)));
typedef float v8f __attribute__((ext_vector_type(8)));

// ---------------- Stage 1: sin/cos tables + f1 column --------------------
__global__ void __launch_bounds__(256)
kuramoto_stage1(const float* __restrict__ t,
                const float* __restrict__ y,
                const float* __restrict__ u,
                float* __restrict__ sinP,
                float* __restrict__ cosP,
                float* __restrict__ out) {
  const int b   = blockIdx.x;    // 0..31
  const int tid = threadIdx.x;   // 0..255
  const int index = (int)floorf(t[0] / DT_);
  const float* __restrict__ urow = u + ((size_t)index * BATCH + b) * DIM;
  const float* __restrict__ yrow = y + (size_t)b * OUTW;   // phases = y[:, :DIM]

  float acc = 0.0f;
  #pragma unroll
  for (int j = tid; j < DIM; j += 256) {
    float p = yrow[j];
    float s = sinf(p);
    float c = cosf(p);
    sinP[(size_t)b * DIM + j] = s;
    cosP[(size_t)b * DIM + j] = c;
    float f = urow[j];
    acc += f * f;
  }

  __shared__ float red[256];
  red[tid] = acc;
  __syncthreads();
  for (int s = 128; s > 0; s >>= 1) {
    if (tid < s) red[tid] += red[tid + s];
    __syncthreads();
  }
  if (tid == 0) out[(size_t)b * OUTW + DIM] = red[0];   // f1[b]
}

// ---------------- Stage 2: fused dual GEMM (WMMA f32 16x16x4) ------------
// One wave per 16x16 output tile of S^T / C^T.
// grid.x = (DIM/16) * (BATCH/16) = 64 * 2 = 128 blocks of 32 threads.
__global__ void __launch_bounds__(32)
kuramoto_stage2(const float* __restrict__ t,
                const float* __restrict__ u,
                const float* __restrict__ A,
                const float* __restrict__ freqs,
                const float* __restrict__ sinP,
                const float* __restrict__ cosP,
                float* __restrict__ out) {
  const int tile = blockIdx.x;
  const int i0 = (tile >> 1) << 4;       // M tile base (A rows / dim index)
  const int n0 = (tile & 1) << 4;        // N tile base (batch index)
  const int lane = threadIdx.x;          // 0..31, full wave (EXEC all 1s)

  // fp32 WMMA operand layout (16x4 A-tile / 4x16 B-tile):
  //   lanes 0-15 : M(or N)=lane,    VGPR0->K=0, VGPR1->K=1
  //   lanes16-31 : M(or N)=lane-16, VGPR0->K=2, VGPR1->K=3
  const int mn = lane & 15;
  const int kb = (lane >> 4) << 1;       // 0 or 2

  const float* __restrict__ arow = A    + (size_t)(i0 + mn) * DIM + kb;
  const float* __restrict__ srow = sinP + (size_t)(n0 + mn) * DIM + kb;
  const float* __restrict__ crow = cosP + (size_t)(n0 + mn) * DIM + kb;

  v8f accS = {};
  v8f accC = {};

  #pragma unroll 4
  for (int j = 0; j < DIM; j += 4) {
    v2f a  = *(const v2f*)(arow + j);
    v2f bs = *(const v2f*)(srow + j);
    v2f bc = *(const v2f*)(crow + j);
    // D = A x B + C   (8-arg form: neg_a, A, neg_b, B, c_mod, C, reuse_a, reuse_b)
    accS = __builtin_amdgcn_wmma_f32_16x16x4_f32(false, a, false, bs,
                                                 (short)0, accS, false, false);
    accC = __builtin_amdgcn_wmma_f32_16x16x4_f32(false, a, false, bc,
                                                 (short)0, accC, false, false);
  }

  // Epilogue: C/D layout -> VGPR r holds (M = r + 8*(lane>=16), N = lane&15).
  const int index = (int)floorf(t[0] / DT_);
  const int b  = n0 + (lane & 15);
  const int mo = (lane >> 4) << 3;       // 0 or 8
  const float* __restrict__ urow = u + ((size_t)index * BATCH + b) * DIM;
  const float* __restrict__ sprow = sinP + (size_t)b * DIM;
  const float* __restrict__ cprow = cosP + (size_t)b * DIM;

  #pragma unroll
  for (int r = 0; r < 8; ++r) {
    const int i = i0 + mo + r;
    const float S = accS[r];
    const float C = accC[r];
    const float coupling = cprow[i] * S - sprow[i] * C;
    const float force = urow[i] * coupling * (1.0f / (float)DIM) + freqs[i];
    out[(size_t)b * OUTW + i] = force;
  }
}

// ---------------------------- Launch -------------------------------------
extern "C" void kernel_launch(void* const* d_in, const int* in_sizes, int n_in,
                              void* d_out, int out_size, void* d_ws, size_t ws_size,
                              hipStream_t stream) {
  (void)in_sizes; (void)n_in; (void)out_size; (void)ws_size;
  const float* t     = (const float*)d_in[0];
  const float* y     = (const float*)d_in[1];
  const float* u     = (const float*)d_in[2];
  const float* A     = (const float*)d_in[3];
  const float* freqs = (const float*)d_in[4];
  float*       out   = (float*)d_out;

  float* sinP = (float*)d_ws;                                   // 32*1024 f32
  float* cosP = (float*)((char*)d_ws + (size_t)BATCH * DIM * 4); // next 128KB

  kuramoto_stage1<<<BATCH, 256, 0, stream>>>(t, y, u, sinP, cosP, out);

  const int tiles = (DIM / 16) * (BATCH / 16);   // 128 waves
  kuramoto_stage2<<<tiles, 32, 0, stream>>>(t, u, A, freqs, sinP, cosP, out);
}